// Decoder_1786706395198
// MI455X (gfx1250) — compile-verified
//
#include <hip/hip_runtime.h>
#include <math.h>

// ---------------------------------------------------------------------------
// Tacotron-style decoder for MI455X (gfx1250, wave32, WMMA f32 16x16x4).
// Every matmul AND every conv1d (as sum of row-shifted GEMMs, with 'SAME'
// masking and fused maxpool) runs through the WMMA f32 pipe; sequential GRU
// scans run as persistent single-workgroup WMMA kernels; the two directions
// of the bidirectional GRU run concurrently as a grid=2 launch.
// ---------------------------------------------------------------------------

typedef __attribute__((ext_vector_type(2))) float v2f;
typedef __attribute__((ext_vector_type(8))) float v8f;

#define ACT_NONE 0
#define ACT_RELU 1
#define ACT_SIGMOID 2

__device__ __forceinline__ float act_apply(float v, int ACT) {
  if (ACT == ACT_RELU) return fmaxf(v, 0.f);
  if (ACT == ACT_SIGMOID) return 1.f / (1.f + __expf(-v));
  return v;
}

// ---------------------------------------------------------------------------
// WMMA GEMM: C[M,N] = act( A' @ B (+ bias) (+ C_old if ACCUM) )
// B is [K,N] row-major (TRANSB=0) or [N,K] row-major (TRANSB=1).
// Conv support: if period>0, A row m is read from row m+rowOff, masked to
// zero when (m%period)+rowOff falls outside [0,period) ('SAME' padding along
// time, per-sequence). POOL=1 additionally fuses max(x[t], x[t+1]) (width-2
// stride-1 maxpool, pad (0,1)) into the A read.
// Block: 256 threads = 8 waves (2x4), block tile 32(M) x 64(N), K chunk 32.
// ---------------------------------------------------------------------------
template <int ACT, int TRANSB, int ACCUM, int POOL>
__global__ __launch_bounds__(256) void gemm_wmma(
    const float* __restrict__ A, const float* __restrict__ B,
    const float* __restrict__ bias, float* __restrict__ C,
    int M, int N, int K, int lda, int ldb, int ldc,
    long strideA, long strideB, long strideC, int rowOff, int period) {
  __shared__ float sA[32][33];
  __shared__ float sB[32][68];

  const int batch = blockIdx.z;
  A += (long)batch * strideA;
  B += (long)batch * strideB;
  C += (long)batch * strideC;

  const int bm = blockIdx.y * 32;
  const int bn = blockIdx.x * 64;
  const int tid = threadIdx.x;
  const int wave = tid >> 5;
  const int lane = tid & 31;
  const int wm = wave >> 2;   // 0..1
  const int wn = wave & 3;    // 0..3
  const int half = lane >> 4; // 0..1
  const int r = lane & 15;    // 0..15
  const int row0 = bm + wm * 16;
  const int col0 = bn + wn * 16;

  v8f acc = {0.f, 0.f, 0.f, 0.f, 0.f, 0.f, 0.f, 0.f};

  for (int k0 = 0; k0 < K; k0 += 32) {
    // Prefetch next K chunk (global_prefetch_b8 on gfx1250).
    if (k0 + 32 < K) {
      if (tid < 32) {
        int gr = bm + tid;
        if (gr < M) __builtin_prefetch(&A[(long)gr * lda + k0 + 32], 0, 1);
      } else if (tid < 96) {
        int rr = tid - 32;
        if (TRANSB) {
          int gn = bn + rr;
          if (gn < N) __builtin_prefetch(&B[(long)gn * ldb + k0 + 32], 0, 1);
        } else if (rr < 32) {
          int gk = k0 + 32 + rr;
          if (gk < K) __builtin_prefetch(&B[(long)gk * ldb + bn], 0, 1);
        }
      }
    }

    // Stage A tile (32 x 32), zero padded, optional row shift/mask/pool.
    for (int i = tid; i < 32 * 32; i += 256) {
      int rr = i >> 5, cc = i & 31;
      int gr = bm + rr, gc = k0 + cc;
      float v = 0.f;
      if (gr < M && gc < K) {
        if (period) {
          int t = gr % period;
          int tt = t + rowOff;
          if (tt >= 0 && tt < period) {
            long ga = (long)(gr + rowOff);
            v = A[ga * lda + gc];
            if (POOL && (tt + 1) < period) v = fmaxf(v, A[(ga + 1) * lda + gc]);
          }
        } else {
          v = A[(long)gr * lda + gc];
        }
      }
      sA[rr][cc] = v;
    }
    // Stage B tile (32 x 64), zero padded.
    for (int i = tid; i < 32 * 64; i += 256) {
      int rr = i >> 6, cc = i & 63;
      int gk = k0 + rr, gn = bn + cc;
      float v = 0.f;
      if (gk < K && gn < N)
        v = TRANSB ? B[(long)gn * ldb + gk] : B[(long)gk * ldb + gn];
      sB[rr][cc] = v;
    }
    __syncthreads();

    // 8 wmma steps of K=4 each (zeros beyond K contribute nothing).
    for (int k = 0; k < 32; k += 4) {
      v2f a, b;
      a.x = sA[wm * 16 + r][k + 2 * half];
      a.y = sA[wm * 16 + r][k + 2 * half + 1];
      b.x = sB[k + 2 * half][wn * 16 + r];
      b.y = sB[k + 2 * half + 1][wn * 16 + r];
      acc = __builtin_amdgcn_wmma_f32_16x16x4_f32(false, a, false, b,
                                                  (short)0, acc, false, false);
    }
    __syncthreads();
  }

  // Epilogue. C/D layout: lane<16 -> col=lane, rows vgpr+0..7;
  //                        lane>=16 -> col=lane-16, rows vgpr+8..15.
  const int col = col0 + r;
  if (col < N) {
    const float bv = bias ? bias[col] : 0.f;
    for (int rr = 0; rr < 8; ++rr) {
      int row = row0 + rr + 8 * half;
      if (row < M) {
        float v = acc[rr] + bv;
        if (ACCUM) v += C[(long)row * ldc + col];
        v = act_apply(v, ACT);
        C[(long)row * ldc + col] = v;
      }
    }
  }
}

// ---------------------------------------------------------------------------
// Persistent GRU scan body (torch GRU math), batch=16, hidden H.
// xw: [16, T, 3H] precomputed input projections (x @ wih.T + bih)
// whh: [3H, H], bhh: [3H], ys: [16, T, H]. 1024 threads = 32 waves.
// sh: LDS [16*H] hidden state; shw: LDS [16*3H] gate pre-activations.
// ---------------------------------------------------------------------------
template <int H>
__device__ __forceinline__ void gru_scan_body(
    const float* __restrict__ xw, const float* __restrict__ whh,
    const float* __restrict__ bhh, float* __restrict__ ys, int T, int reverse,
    float* sh, float* shw) {
  const int tid = threadIdx.x;
  const int wave = tid >> 5;
  const int lane = tid & 31;
  const int half = lane >> 4;
  const int r = lane & 15;
  const int NT = (3 * H) / 16; // 16x16 output tiles of hw

  for (int i = tid; i < 16 * H; i += 1024) sh[i] = 0.f;
  __syncthreads();

  for (int s = 0; s < T; ++s) {
    const int t = reverse ? (T - 1 - s) : s;

    // hw = h @ whh.T + bhh  -> shw[16][3H]
    for (int tile = wave; tile < NT; tile += 32) {
      const int j = tile * 16 + r; // output column (0..3H-1)
      const float* wrow = whh + (long)j * H;
      v8f acc = {0.f, 0.f, 0.f, 0.f, 0.f, 0.f, 0.f, 0.f};
      for (int k = 0; k < H; k += 4) {
        v2f a, b;
        a.x = sh[r * H + k + 2 * half];
        a.y = sh[r * H + k + 2 * half + 1];
        b.x = wrow[k + 2 * half];
        b.y = wrow[k + 2 * half + 1];
        acc = __builtin_amdgcn_wmma_f32_16x16x4_f32(
            false, a, false, b, (short)0, acc, false, false);
      }
      const float bv = bhh[j];
      for (int rr = 0; rr < 8; ++rr)
        shw[(rr + 8 * half) * (3 * H) + j] = acc[rr] + bv;
    }
    __syncthreads();

    // Gate math + state update + output write.
    for (int i = tid; i < 16 * H; i += 1024) {
      const int n = i / H, j = i % H;
      const float* xr = xw + ((long)n * T + t) * (3 * H);
      const float* hw = shw + (long)n * (3 * H);
      float rg = 1.f / (1.f + __expf(-(xr[j] + hw[j])));
      float zg = 1.f / (1.f + __expf(-(xr[H + j] + hw[H + j])));
      float cg = tanhf(xr[2 * H + j] + rg * hw[2 * H + j]);
      float hn = (1.f - zg) * cg + zg * sh[i];
      sh[i] = hn;
      ys[((long)n * T + t) * H + j] = hn;
    }
    __syncthreads();
  }
}

template <int H>
__global__ __launch_bounds__(1024) void gru_scan(
    const float* __restrict__ xw, const float* __restrict__ whh,
    const float* __restrict__ bhh, float* __restrict__ ys, int T,
    int reverse) {
  __shared__ float sh[16 * H];
  __shared__ float shw[16 * 3 * H];
  gru_scan_body<H>(xw, whh, bhh, ys, T, reverse, sh, shw);
}

// Both directions of a bidirectional GRU, one workgroup each (grid = 2),
// running concurrently on two WGPs. Block 0: forward; block 1: backward.
template <int H>
__global__ __launch_bounds__(1024) void bigru_scan2(
    const float* __restrict__ xwf, const float* __restrict__ whhf,
    const float* __restrict__ bhhf, float* __restrict__ ysf,
    const float* __restrict__ xwb, const float* __restrict__ whhb,
    const float* __restrict__ bhhb, float* __restrict__ ysb, int T) {
  __shared__ float sh[16 * H];
  __shared__ float shw[16 * 3 * H];
  const int dir = blockIdx.x;
  gru_scan_body<H>(dir ? xwb : xwf, dir ? whhb : whhf, dir ? bhhb : bhhf,
                   dir ? ysb : ysf, T, dir, sh, shw);
}

// ---------------------------------------------------------------------------
// Bahdanau attention scores + softmax, fused. One block per (n,t).
// wm: [16,128,256], uo: [16,160,256], v: [256], attn: [16,160,128].
// ---------------------------------------------------------------------------
__global__ __launch_bounds__(128) void attn_scores(
    const float* __restrict__ wm, const float* __restrict__ uo,
    const float* __restrict__ vvec, float* __restrict__ attn) {
  const int n = blockIdx.x / 160;
  const int t = blockIdx.x % 160;
  __shared__ float su[256];
  __shared__ float sred[128];
  for (int i = threadIdx.x; i < 256; i += 128)
    su[i] = uo[((long)n * 160 + t) * 256 + i];
  __syncthreads();

  const int x = threadIdx.x; // 0..127 over Tx
  const float* wrow = wm + ((long)n * 128 + x) * 256;
  float s = 0.f;
  for (int e = 0; e < 256; ++e) s += tanhf(wrow[e] + su[e]) * vvec[e];

  sred[x] = s;
  __syncthreads();
  for (int st = 64; st > 0; st >>= 1) {
    if (x < st) sred[x] = fmaxf(sred[x], sred[x + st]);
    __syncthreads();
  }
  const float mx = sred[0];
  __syncthreads();
  const float ex = __expf(s - mx);
  sred[x] = ex;
  __syncthreads();
  for (int st = 64; st > 0; st >>= 1) {
    if (x < st) sred[x] += sred[x + st];
    __syncthreads();
  }
  attn[((long)n * 160 + t) * 128 + x] = ex / sred[0];
}

// BatchNorm (training-mode batch stats) over rows, one block per channel.
__global__ __launch_bounds__(256) void bn_stats(
    const float* __restrict__ x, float* __restrict__ mean,
    float* __restrict__ inv, int rows, int ld, int cbase) {
  const int c = cbase + blockIdx.x;
  __shared__ float ssum[256];
  __shared__ float ssq[256];
  float s = 0.f, q = 0.f;
  for (int rIdx = threadIdx.x; rIdx < rows; rIdx += 256) {
    float v = x[(long)rIdx * ld + c];
    s += v;
    q += v * v;
  }
  ssum[threadIdx.x] = s;
  ssq[threadIdx.x] = q;
  __syncthreads();
  for (int st = 128; st > 0; st >>= 1) {
    if (threadIdx.x < st) {
      ssum[threadIdx.x] += ssum[threadIdx.x + st];
      ssq[threadIdx.x] += ssq[threadIdx.x + st];
    }
    __syncthreads();
  }
  if (threadIdx.x == 0) {
    float m = ssum[0] / (float)rows;
    float var = ssq[0] / (float)rows - m * m;
    mean[blockIdx.x] = m;
    inv[blockIdx.x] = rsqrtf(var + 1e-5f);
  }
}

__global__ __launch_bounds__(256) void bn_apply(
    float* __restrict__ x, const float* __restrict__ mean,
    const float* __restrict__ inv, const float* __restrict__ g,
    const float* __restrict__ b, long rows, int ld, int cbase, int C,
    int relu) {
  long idx = (long)blockIdx.x * blockDim.x + threadIdx.x;
  if (idx >= rows * (long)C) return;
  const int c = (int)(idx % C);
  const long row = idx / C;
  float* p = x + row * (long)ld + cbase + c;
  float v = (*p - mean[c]) * inv[c] * g[c] + b[c];
  if (relu) v = fmaxf(v, 0.f);
  *p = v;
}

__global__ __launch_bounds__(256) void ew_add(float* __restrict__ y,
                                              const float* __restrict__ x,
                                              long n) {
  long i = (long)blockIdx.x * blockDim.x + threadIdx.x;
  if (i < n) y[i] += x[i];
}

__global__ __launch_bounds__(256) void hw_combine(float* __restrict__ h,
                                                  const float* __restrict__ Hh,
                                                  const float* __restrict__ Tt,
                                                  long n) {
  long i = (long)blockIdx.x * blockDim.x + threadIdx.x;
  if (i < n) {
    float tt = Tt[i];
    h[i] = Hh[i] * tt + h[i] * (1.f - tt);
  }
}

// ---------------------------------------------------------------------------
// Host-side GEMM dispatch.
// ---------------------------------------------------------------------------
static void launch_gemm_full(hipStream_t st, const float* A, const float* B,
                             const float* bias, float* C, int M, int N, int K,
                             int lda, int ldb, int ldc, int act, int transB,
                             int accum, int pool, int rowOff, int period,
                             int batch, long strA, long strB, long strC) {
  dim3 g((unsigned)((N + 63) / 64), (unsigned)((M + 31) / 32),
         (unsigned)batch);
  dim3 b(256, 1, 1);
#define GEMM_CASE(A_, T_, C_, P_)                                            \
  gemm_wmma<A_, T_, C_, P_><<<g, b, 0, st>>>(A, B, bias, C, M, N, K, lda,    \
                                             ldb, ldc, strA, strB, strC,     \
                                             rowOff, period)
  if (pool) { // pooled conv taps: act=none, B not transposed
    if (accum) GEMM_CASE(0, 0, 1, 1);
    else GEMM_CASE(0, 0, 0, 1);
  } else if (!transB && !accum) {
    if (act == 0) GEMM_CASE(0, 0, 0, 0);
    else if (act == 1) GEMM_CASE(1, 0, 0, 0);
    else GEMM_CASE(2, 0, 0, 0);
  } else if (transB && !accum) {
    if (act == 0) GEMM_CASE(0, 1, 0, 0);
    else if (act == 1) GEMM_CASE(1, 1, 0, 0);
    else GEMM_CASE(2, 1, 0, 0);
  } else if (!transB && accum) {
    if (act == 0) GEMM_CASE(0, 0, 1, 0);
    else if (act == 1) GEMM_CASE(1, 0, 1, 0);
    else GEMM_CASE(2, 0, 1, 0);
  } else {
    if (act == 0) GEMM_CASE(0, 1, 1, 0);
    else if (act == 1) GEMM_CASE(1, 1, 1, 0);
    else GEMM_CASE(2, 1, 1, 0);
  }
#undef GEMM_CASE
}

static inline void gemm_basic(hipStream_t st, const float* A, const float* B,
                              const float* bias, float* C, int M, int N,
                              int K, int lda, int ldb, int ldc, int act,
                              int transB, int accum) {
  launch_gemm_full(st, A, B, bias, C, M, N, K, lda, ldb, ldc, act, transB,
                   accum, 0, 0, 0, 1, 0, 0, 0);
}

// One conv tap: C (+)= shift(A, rowOff) @ B, rows masked to [0,period).
static inline void gemm_conv(hipStream_t st, const float* A, const float* B,
                             float* C, int M, int N, int K, int lda, int ldb,
                             int ldc, int accum, int pool, int rowOff,
                             int period) {
  launch_gemm_full(st, A, B, nullptr, C, M, N, K, lda, ldb, ldc, ACT_NONE, 0,
                   accum, pool, rowOff, period, 1, 0, 0, 0);
}

static inline dim3 grid1(long n) { return dim3((unsigned)((n + 255) / 256)); }

// ---------------------------------------------------------------------------
extern "C" void kernel_launch(void* const* d_in, const int* in_sizes, int n_in,
                              void* d_out, int out_size, void* d_ws,
                              size_t ws_size, hipStream_t stream) {
  (void)in_sizes; (void)n_in; (void)out_size; (void)ws_size;
#define F(i) ((const float*)d_in[i])
  // Input order (setup_inputs dict order, nested dicts/lists depth-first):
  const float* inputs = F(0);                         // [16,160,80]
  const float* memory = F(1);                         // [16,128,256]
  const float *pre_w1 = F(2), *pre_b1 = F(3), *pre_w2 = F(4), *pre_b2 = F(5);
  const float *ag_wih = F(6), *ag_whh = F(7), *ag_bih = F(8), *ag_bhh = F(9);
  const float *at_W = F(10), *at_U = F(11), *at_v = F(12);
  const float *pj_w = F(13), *pj_b = F(14);
  const float *g1_wih = F(15), *g1_whh = F(16), *g1_bih = F(17),
              *g1_bhh = F(18);
  const float *g2_wih = F(19), *g2_whh = F(20), *g2_bih = F(21),
              *g2_bhh = F(22);
  const float *fc1_w = F(23), *fc1_b = F(24);
  // bank[k]: w=F(25+3k) g=F(26+3k) be=F(27+3k), k=0..7
  const float *p1_w = F(49), *p1_g = F(50), *p1_be = F(51);
  const float *p2_w = F(52), *p2_g = F(53), *p2_be = F(54);
  const float *phw_w = F(55), *phw_b = F(56);
  // hw[i]: w1=F(57+4i) b1=F(58+4i) w2=F(59+4i) b2=F(60+4i), i=0..3
  const float *bf_wih = F(73), *bf_whh = F(74), *bf_bih = F(75),
              *bf_bhh = F(76);
  const float *bb_wih = F(77), *bb_whh = F(78), *bb_bih = F(79),
              *bb_bhh = F(80);
  const float *fc2_w = F(81), *fc2_b = F(82);

  // Output regions: mels [16,160,400] | mags [16,800,1025] | attn [16,160,128]
  float* mels_out = (float*)d_out;
  float* mags_out = mels_out + (size_t)16 * 160 * 400;
  float* attn_out = mags_out + (size_t)16 * 800 * 1025;

  float* ws = (float*)d_ws;
  size_t off = 0;
  auto alloc = [&](size_t n) {
    float* p = ws + off;
    off += (n + 63) & ~(size_t)63;
    return p;
  };
  float* h1 = alloc((size_t)2560 * 256);   // prenet hidden
  float* xb = alloc((size_t)2560 * 128);   // prenet out
  float* xw = alloc((size_t)2560 * 768);   // GRU input projections (reused)
  float* ao = alloc((size_t)2560 * 256);   // attention RNN out
  float* wm = alloc((size_t)2048 * 256);
  float* uo = alloc((size_t)2560 * 256);
  float* ctx = alloc((size_t)2560 * 256);
  float* pj = alloc((size_t)2560 * 256);
  float* g1 = alloc((size_t)2560 * 256);
  float* g2 = alloc((size_t)2560 * 256);
  float* bank = alloc((size_t)12800 * 1024);
  float* c1 = alloc((size_t)12800 * 256);
  float* c2 = alloc((size_t)12800 * 80);
  float* hh = alloc((size_t)12800 * 128);
  float* t1 = alloc((size_t)12800 * 128);
  float* t2 = alloc((size_t)12800 * 128);
  float* bxwf = alloc((size_t)12800 * 384);
  float* bxwb = alloc((size_t)12800 * 384);
  float* fb = alloc((size_t)12800 * 128);
  float* bb = alloc((size_t)12800 * 128);
  float* bmean = alloc(1024);
  float* binv = alloc(1024);

  // 1) PreNet.
  gemm_basic(stream, inputs, pre_w1, pre_b1, h1, 2560, 256, 80, 80, 256, 256,
             ACT_RELU, 0, 0);
  gemm_basic(stream, h1, pre_w2, pre_b2, xb, 2560, 128, 256, 256, 128, 128,
             ACT_RELU, 0, 0);

  // 2) Attention GRU (H=256): xw = x @ wih.T + bih, then persistent scan.
  //    wm = memory @ W is independent of the scan; issue it first so its
  //    blocks can overlap the single-WGP scan.
  gemm_basic(stream, xb, ag_wih, ag_bih, xw, 2560, 768, 128, 128, 128, 768,
             ACT_NONE, 1, 0);
  gemm_basic(stream, memory, at_W, nullptr, wm, 2048, 256, 256, 256, 256, 256,
             ACT_NONE, 0, 0);
  gru_scan<256><<<dim3(1), dim3(1024), 0, stream>>>(xw, ag_whh, ag_bhh, ao,
                                                    160, 0);

  // 3) Bahdanau attention.
  gemm_basic(stream, ao, at_U, nullptr, uo, 2560, 256, 256, 256, 256, 256,
             ACT_NONE, 0, 0);
  attn_scores<<<dim3(2560), dim3(128), 0, stream>>>(wm, uo, at_v, attn_out);
  // ctx = attn @ memory (batched over n=16).
  launch_gemm_full(stream, attn_out, memory, nullptr, ctx, 160, 256, 128, 128,
                   256, 256, ACT_NONE, 0, 0, 0, 0, 0, 16, (long)160 * 128,
                   (long)128 * 256, (long)160 * 256);

  // 4) proj = concat(ctx, ao) @ pj_w + pj_b  (split-K as two GEMMs).
  gemm_basic(stream, ctx, pj_w, pj_b, pj, 2560, 256, 256, 256, 256, 256,
             ACT_NONE, 0, 0);
  gemm_basic(stream, ao, pj_w + 256 * 256, nullptr, pj, 2560, 256, 256, 256,
             256, 256, ACT_NONE, 0, 1);

  // 5) gru1 + residual.
  gemm_basic(stream, pj, g1_wih, g1_bih, xw, 2560, 768, 256, 256, 256, 768,
             ACT_NONE, 1, 0);
  gru_scan<256><<<dim3(1), dim3(1024), 0, stream>>>(xw, g1_whh, g1_bhh, g1,
                                                    160, 0);
  ew_add<<<grid1((long)2560 * 256), 256, 0, stream>>>(g1, pj,
                                                      (long)2560 * 256);
  // 6) gru2 + residual.
  gemm_basic(stream, g1, g2_wih, g2_bih, xw, 2560, 768, 256, 256, 256, 768,
             ACT_NONE, 1, 0);
  gru_scan<256><<<dim3(1), dim3(1024), 0, stream>>>(xw, g2_whh, g2_bhh, g2,
                                                    160, 0);
  ew_add<<<grid1((long)2560 * 256), 256, 0, stream>>>(g2, g1,
                                                      (long)2560 * 256);

  // 7) fc1 -> mels (d_out region; viewed as y[16,800,80] below).
  gemm_basic(stream, g2, fc1_w, fc1_b, mels_out, 2560, 400, 256, 256, 400,
             400, ACT_NONE, 0, 0);

  // 8) Conv bank (widths 1..8) as WMMA tap-GEMMs, then BN + ReLU.
  //    Tap dk of conv k: bank[:, k*128 + :] += shift(y, dk - k/2) @ w_k[dk].
  for (int k = 0; k < 8; ++k) {
    const int KW = k + 1, padlo = k / 2;
    for (int dk = 0; dk < KW; ++dk) {
      gemm_conv(stream, mels_out, F(25 + 3 * k) + (size_t)dk * 80 * 128,
                bank + k * 128, 12800, 128, 80, 80, 128, 1024, dk > 0, 0,
                dk - padlo, 800);
    }
  }
  for (int k = 0; k < 8; ++k) {
    bn_stats<<<dim3(128), dim3(256), 0, stream>>>(bank, bmean, binv, 12800,
                                                  1024, k * 128);
    bn_apply<<<grid1((long)12800 * 128), 256, 0, stream>>>(
        bank, bmean, binv, F(26 + 3 * k), F(27 + 3 * k), 12800, 1024, k * 128,
        128, 1);
  }

  // 9) proj1 conv as 3 WMMA tap-GEMMs with maxpool fused into A reads.
  for (int dk = 0; dk < 3; ++dk) {
    gemm_conv(stream, bank, p1_w + (size_t)dk * 1024 * 256, c1, 12800, 256,
              1024, 1024, 256, 256, dk > 0, /*pool=*/1, dk - 1, 800);
  }
  bn_stats<<<dim3(256), dim3(256), 0, stream>>>(c1, bmean, binv, 12800, 256,
                                                0);
  bn_apply<<<grid1((long)12800 * 256), 256, 0, stream>>>(
      c1, bmean, binv, p1_g, p1_be, 12800, 256, 0, 256, 1);

  // 10) proj2 conv as 3 WMMA tap-GEMMs + BN (no relu) + residual with y.
  for (int dk = 0; dk < 3; ++dk) {
    gemm_conv(stream, c1, p2_w + (size_t)dk * 256 * 80, c2, 12800, 80, 256,
              256, 80, 80, dk > 0, 0, dk - 1, 800);
  }
  bn_stats<<<dim3(80), dim3(256), 0, stream>>>(c2, bmean, binv, 12800, 80, 0);
  bn_apply<<<grid1((long)12800 * 80), 256, 0, stream>>>(
      c2, bmean, binv, p2_g, p2_be, 12800, 80, 0, 80, 0);
  ew_add<<<grid1((long)12800 * 80), 256, 0, stream>>>(c2, mels_out,
                                                      (long)12800 * 80);

  // 11) Pre-highway projection to E/2.
  gemm_basic(stream, c2, phw_w, phw_b, hh, 12800, 128, 80, 80, 128, 128,
             ACT_NONE, 0, 0);

  // 12) Highway x4.
  for (int i = 0; i < 4; ++i) {
    gemm_basic(stream, hh, F(57 + 4 * i), F(58 + 4 * i), t1, 12800, 128, 128,
               128, 128, 128, ACT_RELU, 0, 0);
    gemm_basic(stream, hh, F(59 + 4 * i), F(60 + 4 * i), t2, 12800, 128, 128,
               128, 128, 128, ACT_SIGMOID, 0, 0);
    hw_combine<<<grid1((long)12800 * 128), 256, 0, stream>>>(
        hh, t1, t2, (long)12800 * 128);
  }

  // 13) Bi-GRU (H=128, T=800): both input projections first, then the two
  //     independent 800-step scans run concurrently (grid = 2, one WGP each).
  gemm_basic(stream, hh, bf_wih, bf_bih, bxwf, 12800, 384, 128, 128, 128, 384,
             ACT_NONE, 1, 0);
  gemm_basic(stream, hh, bb_wih, bb_bih, bxwb, 12800, 384, 128, 128, 128, 384,
             ACT_NONE, 1, 0);
  bigru_scan2<128><<<dim3(2), dim3(1024), 0, stream>>>(
      bxwf, bf_whh, bf_bhh, fb, bxwb, bb_whh, bb_bhh, bb, 800);

  // 14) fc2 -> mags = concat(f, bwd) @ fc2_w + b (split-K as two GEMMs).
  gemm_basic(stream, fb, fc2_w, fc2_b, mags_out, 12800, 1025, 128, 128, 1025,
             1025, ACT_NONE, 0, 0);
  gemm_basic(stream, bb, fc2_w + 128 * 1025, nullptr, mags_out, 12800, 1025,
             128, 128, 1025, 1025, ACT_NONE, 0, 1);
#undef F
}